// FilterDetections_30940944401107
// MI455X (gfx1250) — compile-verified
//
#include <hip/hip_runtime.h>
#include <stdint.h>

#define B_IMG   4
#define N_ANCH  49104
#define N_CLS   90
#define MAX_DET 100
#define CAP_PER_IMG (N_CLS * MAX_DET)   /* 9000 */
#define NEGV     (-1e30f)
#define NEG_HALF (-5e29f)
#define SCORE_THR 0.01f
#define IOU_THR   0.1f

// TDM builtins only exist in the device pass; hard-fail there if missing so a
// successful compile PROVES the tensor path is in the gfx1250 binary.
#if defined(__HIP_DEVICE_COMPILE__)
#  if !__has_builtin(__builtin_amdgcn_tensor_load_to_lds) || !__has_builtin(__builtin_amdgcn_s_wait_tensorcnt)
#    error "CDNA5 TDM builtins not available in device pass"
#  endif
#  define TDM_AVAILABLE 1
#else
#  define TDM_AVAILABLE 0   /* host pass: parse-only stubs */
#endif

typedef unsigned int u32x4 __attribute__((ext_vector_type(4)));
typedef int          i32x4 __attribute__((ext_vector_type(4)));
typedef int          i32x8 __attribute__((ext_vector_type(8)));

// Issue one TDM 2-D tile load (4-byte elements) from global to LDS.
// tile walks tile_d0 contiguous elems per row, tile_d1 rows, rows separated by
// stride0 elements in global memory; lands contiguous in LDS at lds_off.
__device__ __forceinline__ void tdm_load_2d(uint32_t lds_off, uint64_t ga,
                                            uint32_t tile_d0, uint32_t tile_d1,
                                            uint32_t tensor_d0, uint32_t tensor_d1,
                                            uint32_t stride0) {
#if TDM_AVAILABLE
  u32x4 g0;
  g0[0] = 1u;                                                // count=1, user mode
  g0[1] = lds_off;                                           // lds_addr (bytes)
  g0[2] = (uint32_t)ga;                                      // global_addr[31:0]
  g0[3] = (uint32_t)((ga >> 32) & 0x01FFFFFFu) | (2u << 30); // addr[56:32] | type=2
  i32x8 g1;
  g1[0] = 0x00020000;                                        // data_size = 4B
  g1[1] = (int)((tensor_d0 & 0xFFFFu) << 16);                // tensor_dim0 lo16
  g1[2] = (int)(((tensor_d0 >> 16) & 0xFFFFu) |              // tensor_dim0 hi16
                ((tensor_d1 & 0xFFFFu) << 16));              // tensor_dim1 lo16
  g1[3] = (int)(((tensor_d1 >> 16) & 0xFFFFu) |              // tensor_dim1 hi16
                ((tile_d0 & 0xFFFFu) << 16));                // tile_dim0
  g1[4] = (int)(tile_d1 & 0xFFFFu);                          // tile_dim1, tile_dim2=0
  g1[5] = (int)stride0;                                      // tensor_dim0_stride lo32
  g1[6] = 0;
  g1[7] = 0;
  i32x4 g2; g2[0] = 1; g2[1] = 1; g2[2] = 0; g2[3] = 0;      // dim2=dim3=1 (unused)
  i32x4 g3; g3[0] = 0; g3[1] = (1 << 16); g3[2] = 0; g3[3] = 0; // tensor_dim4=1
#if defined(__clang_major__) && (__clang_major__ >= 23)
  i32x8 gz = {};
  __builtin_amdgcn_tensor_load_to_lds(g0, g1, g2, g3, gz, 0);
#else
  __builtin_amdgcn_tensor_load_to_lds(g0, g1, g2, g3, 0);
#endif
#else
  (void)lds_off; (void)ga; (void)tile_d0; (void)tile_d1;
  (void)tensor_d0; (void)tensor_d1; (void)stride0;
#endif
}

__device__ __forceinline__ void tdm_wait_all() {
#if TDM_AVAILABLE
  __builtin_amdgcn_s_wait_tensorcnt(0);
#endif
}

// wave32 + cross-wave argmax with first-index tie-break (matches jnp.argmax)
__device__ __forceinline__ void block_argmax(float& v, int& i,
                                             volatile float* red_v,
                                             volatile int* red_i) {
#pragma unroll
  for (int off = 16; off > 0; off >>= 1) {
    float ov = __shfl_xor(v, off, 32);
    int   oi = __shfl_xor(i, off, 32);
    if (ov > v || (ov == v && oi < i)) { v = ov; i = oi; }
  }
  const int w = threadIdx.x >> 5;
  if ((threadIdx.x & 31) == 0) { red_v[w] = v; red_i[w] = i; }
  __syncthreads();
  float fv = red_v[0]; int fi = red_i[0];
#pragma unroll
  for (int k = 1; k < 8; ++k) {
    float ov = red_v[k]; int oi = red_i[k];
    if (ov > fv || (ov == fv && oi < fi)) { fv = ov; fi = oi; }
  }
  v = fv; i = fi;
  __syncthreads();
}

__global__ void init_kernel(int* counts) {
  if (threadIdx.x < B_IMG) counts[threadIdx.x] = 0;
}

// One block per (image b = blockIdx.y, class c = blockIdx.x).
__global__ void __launch_bounds__(256)
nms_kernel(const float4* __restrict__ boxesv, const float* __restrict__ cls,
           uint2* __restrict__ cand, int* __restrict__ counts) {
  extern __shared__ float s[];                 // N_ANCH score slots (192 KB LDS)
  __shared__ float red_v[8];
  __shared__ int   red_i[8];
  __shared__ int   cand_i[MAX_DET];
  __shared__ float cand_s[MAX_DET];
  __shared__ int   base_slot;

  const int tid = threadIdx.x;
  const int c = blockIdx.x;
  const int b = blockIdx.y;
  const float*  col = cls + (size_t)b * N_ANCH * N_CLS + c;  // stride N_CLS floats
  const float4* bx  = boxesv + (size_t)b * N_ANCH;

  // TDM: gather the stride-90 class column (N_ANCH rows x 1 elem, 4B) into LDS.
  if (tid < 32) {
    tdm_load_2d((uint32_t)(uintptr_t)(void*)s, (uint64_t)(uintptr_t)col,
                /*tile_d0=*/1u, /*tile_d1=*/(uint32_t)N_ANCH,
                /*tensor_d0=*/(uint32_t)N_CLS, /*tensor_d1=*/(uint32_t)N_ANCH,
                /*stride0=*/(uint32_t)N_CLS);
    tdm_wait_all();
  }
  __syncthreads();
  for (int n = tid; n < N_ANCH; n += 256) {
    float v = s[n];
    s[n] = (v > SCORE_THR) ? v : NEGV;
  }
  __syncthreads();

  // initial argmax
  float mv = NEGV; int mi = 0;
  for (int n = tid; n < N_ANCH; n += 256) {
    float v = s[n];
    if (v > mv) { mv = v; mi = n; }   // strict > keeps first index per thread
  }
  block_argmax(mv, mi, red_v, red_i);

  int sel = 0;
  while (mv > NEG_HALF) {             // matches ok = s[i] > NEG*0.5
    if (tid == 0) { cand_i[sel] = mi; cand_s[sel] = mv; s[mi] = NEGV; }
    ++sel;
    if (sel == MAX_DET) break;        // 100th pick needs no further suppression
    const float4 bb = bx[mi];
    const float  ba = (bb.z - bb.x) * (bb.w - bb.y);
    __syncthreads();
    // fused suppression + next argmax
    mv = NEGV; mi = 0;
    for (int n = tid; n < N_ANCH; n += 256) {
      float v = s[n];
      if (v > NEG_HALF) {
        const float4 q = bx[n];
        const float iy = fminf(bb.z, q.z) - fmaxf(bb.x, q.x);
        const float ix = fminf(bb.w, q.w) - fmaxf(bb.y, q.y);
        const float inter = fmaxf(iy, 0.f) * fmaxf(ix, 0.f);
        const float qa = (q.z - q.x) * (q.w - q.y);
        if (inter > IOU_THR * (ba + qa - inter + 1e-8f)) { s[n] = NEGV; v = NEGV; }
      }
      if (v > mv) { mv = v; mi = n; }
    }
    block_argmax(mv, mi, red_v, red_i);
  }

  // flush selections to the per-image candidate list
  if (tid == 0) base_slot = atomicAdd(&counts[b], sel);
  __syncthreads();
  const int base = base_slot;
  for (int j = tid; j < sel; j += 256) {
    uint2 p;
    p.x = __float_as_uint(cand_s[j]);
    p.y = (uint32_t)cand_i[j] * (uint32_t)N_CLS + (uint32_t)c;  // flat = anchor*C + class
    cand[(size_t)b * CAP_PER_IMG + base + j] = p;
  }
}

// One block per image: top-100 over <=9000 candidates, score desc / flat idx asc.
__global__ void __launch_bounds__(256)
topk_kernel(const float4* __restrict__ boxesv, const uint2* __restrict__ cand,
            const int* __restrict__ counts, float* __restrict__ out) {
  extern __shared__ unsigned long long key[];   // CAP_PER_IMG keys (72 KB)
  __shared__ unsigned long long red_k[8];
  __shared__ int red_p[8];
  const int tid = threadIdx.x;
  const int b = blockIdx.x;
  int K = counts[b];
  if (K > CAP_PER_IMG) K = CAP_PER_IMG;

  // TDM: contiguous 2*K-dword copy of this image's candidate pairs into LDS.
  if (tid < 32 && K > 0) {
    const uint32_t nd = (uint32_t)(2 * K);       // dwords, <= 18000 (fits 16-bit tile dim)
    tdm_load_2d((uint32_t)(uintptr_t)(void*)key,
                (uint64_t)(uintptr_t)(cand + (size_t)b * CAP_PER_IMG),
                /*tile_d0=*/nd, /*tile_d1=*/1u,
                /*tensor_d0=*/nd, /*tensor_d1=*/1u, /*stride0=*/nd);
    tdm_wait_all();
  }
  __syncthreads();
  // in-place: (score_bits, flat) pair -> 64-bit sort key in the same 8-byte slot
  uint2* pairs = (uint2*)key;
  for (int j = tid; j < K; j += 256) {
    const uint2 p = pairs[j];
    key[j] = ((unsigned long long)p.x << 32) |
             (unsigned long long)(0xFFFFFFFFu - p.y);  // ~idx -> lower idx wins ties
  }
  __syncthreads();

  float* bo = out;                          // boxes  [4,100,4]
  float* so = out + B_IMG * MAX_DET * 4;    // scores [4,100]
  float* lo = so + B_IMG * MAX_DET;         // labels [4,100] (written as float)

  for (int outj = 0; outj < MAX_DET; ++outj) {
    unsigned long long kk = 0ull; int kp = -1;
    for (int j = tid; j < K; j += 256) {
      const unsigned long long t = key[j];
      if (t > kk) { kk = t; kp = j; }
    }
#pragma unroll
    for (int off = 16; off > 0; off >>= 1) {
      const unsigned long long ok = __shfl_xor(kk, off, 32);
      const int op = __shfl_xor(kp, off, 32);
      if (ok > kk) { kk = ok; kp = op; }    // keys unique -> no tie handling needed
    }
    const int w = tid >> 5;
    if ((tid & 31) == 0) { red_k[w] = kk; red_p[w] = kp; }
    __syncthreads();
    unsigned long long fk = red_k[0]; int fp = red_p[0];
#pragma unroll
    for (int k2 = 1; k2 < 8; ++k2) {
      if (red_k[k2] > fk) { fk = red_k[k2]; fp = red_p[k2]; }
    }
    __syncthreads();
    if (tid == 0) {
      const int o4 = (b * MAX_DET + outj) * 4;
      const int o1 = b * MAX_DET + outj;
      if (fk != 0ull) {
        const uint32_t sb   = (uint32_t)(fk >> 32);
        const uint32_t flat = 0xFFFFFFFFu - (uint32_t)(fk & 0xFFFFFFFFull);
        const int anchor = (int)(flat / N_CLS);
        const int label  = (int)(flat % N_CLS);
        const float4 q = boxesv[(size_t)b * N_ANCH + anchor];
        bo[o4 + 0] = q.x; bo[o4 + 1] = q.y; bo[o4 + 2] = q.z; bo[o4 + 3] = q.w;
        so[o1] = __uint_as_float(sb);
        lo[o1] = (float)label;
        key[fp] = 0ull;
      } else {
        bo[o4 + 0] = -1.f; bo[o4 + 1] = -1.f; bo[o4 + 2] = -1.f; bo[o4 + 3] = -1.f;
        so[o1] = -1.f;
        lo[o1] = -1.f;
      }
    }
    __syncthreads();
  }
}

extern "C" void kernel_launch(void* const* d_in, const int* in_sizes, int n_in,
                              void* d_out, int out_size, void* d_ws, size_t ws_size,
                              hipStream_t stream) {
  (void)in_sizes; (void)n_in; (void)out_size; (void)ws_size;
  const float4* boxesv = (const float4*)d_in[0];      // [4,49104,4] f32
  const float*  cls    = (const float*)d_in[1];       // [4,49104,90] f32
  float* out = (float*)d_out;                         // 2400 f32
  int*   counts = (int*)d_ws;                         // 4 ints
  uint2* cand   = (uint2*)((char*)d_ws + 256);        // 4 x 9000 pairs

  init_kernel<<<1, 32, 0, stream>>>(counts);

  const size_t smem1 = (size_t)N_ANCH * sizeof(float);            // 192 KB (CDNA5 WGP LDS)
  (void)hipFuncSetAttribute(reinterpret_cast<const void*>(nms_kernel),
                            hipFuncAttributeMaxDynamicSharedMemorySize, (int)smem1);
  dim3 g1(N_CLS, B_IMG);
  nms_kernel<<<g1, 256, smem1, stream>>>(boxesv, cls, cand, counts);

  const size_t smem2 = (size_t)CAP_PER_IMG * sizeof(unsigned long long);  // 72 KB
  (void)hipFuncSetAttribute(reinterpret_cast<const void*>(topk_kernel),
                            hipFuncAttributeMaxDynamicSharedMemorySize, (int)smem2);
  topk_kernel<<<B_IMG, 256, smem2, stream>>>(boxesv, cand, counts, out);
}